// VanillaPLI_71889162600689
// MI455X (gfx1250) — compile-verified
//
#include <hip/hip_runtime.h>

// y_t = sigmoid(beta_raw) * y_{t-1} + x_t  over (T=128, B=64, N=4096) fp32.
// Blocked-scan: per 16-step block, y_blk = L16 @ x_blk + beta^(i+1)*carry,
// computed with 4 chained V_WMMA_F32_16X16X4_F32 (carry pre-seeded into C).
// Shapes are fixed by the reference, so the 1 MiB row stride is a compile-time
// constant -> all global accesses are vgpr-base + immediate-offset.

typedef __attribute__((ext_vector_type(2))) float v2f;
typedef __attribute__((ext_vector_type(8))) float v8f;

#define T_STEPS  128
#define T_BLOCKS 8                       // 128 / 16
constexpr int NCOLS = 64 * 4096;        // B*N = 262144 floats per time row
constexpr size_t ROW = (size_t)NCOLS;   // row stride in floats (1 MiB in bytes)

__global__ __launch_bounds__(256) void pli_scan_wmma(
    const float* __restrict__ x,
    const float* __restrict__ beta_raw_p,
    float* __restrict__ y)
{
  const int lane  = threadIdx.x & 31;
  const int wave  = threadIdx.x >> 5;
  const int strip = blockIdx.x * 8 + wave;   // one 16-column strip per wave
  const int colBase = strip * 16;

  const int  m   = lane & 15;
  const bool hi  = (lane >= 16);
  const int  khi = hi ? 2 : 0;   // K-half offset for A/B layouts
  const int  rhi = hi ? 8 : 0;   // row-half offset for C/D layout

  // beta = sigmoid(beta_raw); exact small powers via bit decomposition.
  const float braw = beta_raw_p[0];
  const float beta = 1.0f / (1.0f + __expf(-braw));
  const float b2 = beta * beta, b4 = b2 * b2, b8 = b4 * b4;
  auto bpow = [&](int e) -> float {   // beta^e for e in [0,15]
    float r = 1.0f;
    r *= (e & 1) ? beta : 1.0f;
    r *= (e & 2) ? b2   : 1.0f;
    r *= (e & 4) ? b4   : 1.0f;
    r *= (e & 8) ? b8   : 1.0f;
    return r;
  };

  // A = L16 (lower-triangular, L[m][s] = beta^(m-s)), as four 16x4 K-slices.
  // A layout (32-bit 16x4): lanes 0-15 -> M=lane, K={0,1}; lanes 16-31 -> K={2,3}.
  v2f a[4];
#pragma unroll
  for (int k = 0; k < 4; ++k) {
    const int s0 = 4 * k + khi;
    const int e0 = m - s0, e1 = m - (s0 + 1);
    a[k].x = (e0 < 0) ? 0.0f : bpow(e0);
    a[k].y = (e1 < 0) ? 0.0f : bpow(e1);
  }

  // C-seed factors: c[r] holds rows {r, r+8}; seed = beta^(row+1) * carry.
  float f[8];
  const float mhi = hi ? b8 : 1.0f;
#pragma unroll
  for (int r = 0; r < 8; ++r) f[r] = bpow(r + 1) * mhi;

  // Per-lane base pointers with the lane's half-offsets folded in once.
  // Remaining per-access offsets are compile-time constants <= 7 MiB,
  // which fit the signed 24-bit global IOFFSET.
  const float* xlo = x + (size_t)colBase + (size_t)m + (size_t)khi * ROW; // rows 0..7  (+khi)
  const float* xhi = xlo + 8 * ROW;                                      // rows 8..15 (+khi)
  float*       yb  = y + (size_t)colBase + (size_t)m + (size_t)rhi * ROW;

  float carry = 0.0f;

  for (int blk = 0; blk < T_BLOCKS; ++blk) {
    // B tile (4x16 per WMMA step): lane holds rows {4k+khi, 4k+khi+1}, col = m.
    float bv0[4], bv1[4];
    bv0[0] = xlo[0];        bv1[0] = xlo[ROW];         // rows khi+0,  khi+1
    bv0[1] = xlo[4 * ROW];  bv1[1] = xlo[5 * ROW];     // rows khi+4,  khi+5
    bv0[2] = xhi[0];        bv1[2] = xhi[ROW];         // rows khi+8,  khi+9
    bv0[3] = xhi[4 * ROW];  bv1[3] = xhi[5 * ROW];     // rows khi+12, khi+13

    // Seed accumulator with carry contribution: C[i][n] = beta^(i+1) * carry[n].
    v8f c;
#pragma unroll
    for (int r = 0; r < 8; ++r) c[r] = f[r] * carry;

    // y_blk = L16 @ x_blk + C  via 4 chained K=4 f32 WMMAs.
#pragma unroll
    for (int k = 0; k < 4; ++k) {
      v2f b; b.x = bv0[k]; b.y = bv1[k];
      c = __builtin_amdgcn_wmma_f32_16x16x4_f32(
              /*neg_a=*/false, a[k], /*neg_b=*/false, b,
              /*c_mod=*/(short)0, c, /*reuse_a=*/false, /*reuse_b=*/false);
    }

    // Store D: VGPR r -> row (r + rhi), col m (64B-contiguous per half-wave).
#pragma unroll
    for (int r = 0; r < 8; ++r) {
      yb[(size_t)r * ROW] = c[r];
    }

    // carry[n] = y[row 15][n], held in c[7] lanes 16..31; broadcast to all lanes.
    carry = __int_as_float(
        __builtin_amdgcn_ds_bpermute((16 | m) << 2, __float_as_int(c[7])));

    // Advance one 16-row time block (16 MiB) — three 64-bit adds total.
    xlo += 16 * ROW;
    xhi += 16 * ROW;
    yb  += 16 * ROW;
  }
}

extern "C" void kernel_launch(void* const* d_in, const int* in_sizes, int n_in,
                              void* d_out, int out_size, void* d_ws, size_t ws_size,
                              hipStream_t stream) {
  const float* x    = (const float*)d_in[0];
  const float* braw = (const float*)d_in[1];
  float*       yout = (float*)d_out;

  // Fixed shapes per reference: strips = NCOLS/16 = 16384 waves,
  // 8 waves (256 threads) per block -> 2048 blocks, no partial waves
  // (EXEC all-ones, as WMMA requires).
  const int strips = NCOLS / 16;
  const int blocks = strips / 8;

  pli_scan_wmma<<<blocks, 256, 0, stream>>>(x, braw, yout);
}